// Dfc_83580063580960
// MI455X (gfx1250) — compile-verified
//
#include <hip/hip_runtime.h>

#define VOCAB 100000
#define DIMS 64

// ---------------------------------------------------------------------------
// Kernel 1: Wt[v][d] = W[d][v] + b[d]   (transpose with bias folded in)
// W: [DIMS, VOCAB] row-major.  Tiled through LDS, conflict-free (64x65 pad).
// Fast float4 path for full tiles; guarded scalar path for the last tile.
// ---------------------------------------------------------------------------
__global__ void transpose_bias_kernel(const float* __restrict__ W,
                                      const float* __restrict__ b,
                                      float* __restrict__ Wt) {
    __shared__ float tile[64][65];
    const int v0  = blockIdx.x * 64;
    const int tid = threadIdx.x;        // 256 threads

    if (v0 + 64 <= VOCAB) {
        // ---- full tile: vectorized, no guards ----
        // Load: float4 along v. thread -> (d = tid>>4 in 0..15, v4 = (tid&15)*4)
        const int dL = tid >> 4;
        const int v4 = (tid & 15) * 4;
        #pragma unroll
        for (int i = 0; i < 4; ++i) {
            int d = i * 16 + dL;
            float4 r = *(const float4*)(W + (size_t)d * VOCAB + v0 + v4);
            tile[d][v4 + 0] = r.x;
            tile[d][v4 + 1] = r.y;
            tile[d][v4 + 2] = r.z;
            tile[d][v4 + 3] = r.w;
        }
        __syncthreads();
        // Store: float4 along d. thread -> (d4 = (tid&15)*4, v = i*16 + (tid>>4))
        const int d4 = (tid & 15) * 4;
        const int vL = tid >> 4;
        const float4 bias = *(const float4*)(b + d4);
        #pragma unroll
        for (int i = 0; i < 4; ++i) {
            int v = i * 16 + vL;
            float4 o;
            o.x = tile[d4 + 0][v] + bias.x;
            o.y = tile[d4 + 1][v] + bias.y;
            o.z = tile[d4 + 2][v] + bias.z;
            o.w = tile[d4 + 3][v] + bias.w;
            *(float4*)(Wt + (size_t)(v0 + v) * DIMS + d4) = o;
        }
    } else {
        // ---- partial last tile: scalar, guarded ----
        const int dl = tid >> 6;        // 0..3
        const int vl = tid & 63;        // 0..63
        #pragma unroll
        for (int i = 0; i < 16; ++i) {
            int d = i * 4 + dl;
            int v = v0 + vl;
            tile[d][vl] = (v < VOCAB) ? W[(size_t)d * VOCAB + v] : 0.0f;
        }
        __syncthreads();
        const int dd = tid & 63;
        const int vb = tid >> 6;
        const float bias = b[dd];
        #pragma unroll
        for (int i = 0; i < 16; ++i) {
            int v  = i * 4 + vb;
            int gv = v0 + v;
            if (gv < VOCAB) Wt[(size_t)gv * DIMS + dd] = tile[dd][v] + bias;
        }
    }
}

// ---------------------------------------------------------------------------
// Kernel 2: async-DMA gather.  16 lanes per row; each lane moves 16 bytes
// global(L2-resident Wt) -> LDS -> global(out) via the CDNA5 async path.
// Payload never touches VGPRs; completion tracked with ASYNCcnt.
// Load with HT (keep the 25.6 MB table hot in the 192 MB L2); store with NT
// (210 MB streaming output must not thrash L2).
// ---------------------------------------------------------------------------
__global__ void gather_async_kernel(const int* __restrict__ idx,
                                    const float* __restrict__ Wt,
                                    float* __restrict__ out,
                                    int nrows16) {
    __shared__ char lds_buf[256 * 16];  // one 16B slot per thread
    const int tid = threadIdx.x;
    const int row = blockIdx.x * 16 + (tid >> 4);
    if (row >= nrows16) return;
    const int seg = (tid & 15) * 16;    // byte segment within the 256B row

    // 32-bit wave-relative LDS byte address of this lane's slot.
    unsigned lds_off = (unsigned)(unsigned long long)
        (__attribute__((address_space(3))) char*)(&lds_buf[tid * 16]);

    const int v = idx[row];
    const char* src = (const char*)(Wt + (size_t)v   * DIMS) + seg;
    char*       dst = (char*)      (out + (size_t)row * DIMS) + seg;

    // global -> LDS (async, ASYNCcnt++), high-temporal: keep table in L2
    asm volatile("global_load_async_to_lds_b128 %0, %1, off th:TH_LOAD_HT"
                 :: "v"(lds_off), "v"(src) : "memory");
    // wait for the load's data to land in LDS
    asm volatile("s_wait_asynccnt 0x0" ::: "memory");
    // LDS -> global (async, ASYNCcnt++), non-temporal streaming store
    asm volatile("global_store_async_from_lds_b128 %0, %1, off th:TH_STORE_NT"
                 :: "v"(dst), "v"(lds_off) : "memory");
    // drain before wave exit (S_ENDPGM also implies wait-idle; be explicit)
    asm volatile("s_wait_asynccnt 0x0" ::: "memory");
}

// ---------------------------------------------------------------------------
// Tail kernel (rows not covered by the 16-row-granular async kernel).
// Reads the bias-folded table directly.
// ---------------------------------------------------------------------------
__global__ void gather_tail_kernel(const int* __restrict__ idx,
                                   const float* __restrict__ Wt,
                                   float* __restrict__ out,
                                   int nrows, int start) {
    int t   = blockIdx.x * 256 + threadIdx.x;
    int row = start + (t >> 6);
    int d   = t & 63;
    if (row < nrows)
        out[(size_t)row * DIMS + d] = Wt[(size_t)idx[row] * DIMS + d];
}

// ---------------------------------------------------------------------------
// Fallback if workspace can't hold the 25.6 MB transposed table:
// strided direct gather from W (correct, slower).
// ---------------------------------------------------------------------------
__global__ void gather_direct_kernel(const int* __restrict__ idx,
                                     const float* __restrict__ W,
                                     const float* __restrict__ b,
                                     float* __restrict__ out,
                                     int nrows) {
    int t   = blockIdx.x * 256 + threadIdx.x;
    int row = t >> 6;
    int d   = t & 63;
    if (row < nrows) {
        int v = idx[row];
        out[(size_t)row * DIMS + d] = W[(size_t)d * VOCAB + v] + b[d];
    }
}

extern "C" void kernel_launch(void* const* d_in, const int* in_sizes, int n_in,
                              void* d_out, int out_size, void* d_ws, size_t ws_size,
                              hipStream_t stream) {
    const int*   x = (const int*)  d_in[0];   // indices (JAX default: int32)
    const float* W = (const float*)d_in[1];   // [DIMS, VOCAB]
    const float* b = (const float*)d_in[2];   // [DIMS]
    float* out = (float*)d_out;
    const int nrows = in_sizes[0];            // 4096*200 = 819200

    const size_t wt_bytes = (size_t)VOCAB * DIMS * sizeof(float);  // 25.6 MB

    if (ws_size >= wt_bytes) {
        float* Wt = (float*)d_ws;
        // 1) transpose + bias fold (runs every call; deterministic)
        transpose_bias_kernel<<<(VOCAB + 63) / 64, 256, 0, stream>>>(W, b, Wt);

        // 2) async-DMA gather over the 16-row-aligned body
        int body = nrows & ~15;
        if (body > 0)
            gather_async_kernel<<<body / 16, 256, 0, stream>>>(x, Wt, out, body);

        // 3) tail (none for 819200, but stay correct for any size)
        int rem = nrows - body;
        if (rem > 0) {
            int tail_threads = rem * DIMS;
            gather_tail_kernel<<<(tail_threads + 255) / 256, 256, 0, stream>>>(
                x, Wt, out, nrows, body);
        }
    } else {
        long long total = (long long)nrows * DIMS;
        gather_direct_kernel<<<(int)((total + 255) / 256), 256, 0, stream>>>(
            x, W, b, out, nrows);
    }
}